// SAGEStage3_ReduceSum_51994874085795
// MI455X (gfx1250) — compile-verified
//
#include <hip/hip_runtime.h>

typedef float v4f __attribute__((ext_vector_type(4)));

#define FEAT 64
#define THREADS_PER_EDGE 16   // 64 feats / 4 floats per b128

// ---------------------------------------------------------------------------
// Scatter-add (defined FIRST so its disasm shows in the feedback snippet).
// Thread (e, q): one NT global_load_b128 of messages[e, q:q+4], then 4
// relaxed agent-scope f32 atomic adds with discarded results ->
// non-returning GLOBAL_ATOMIC_ADD_F32, resolved at the L2 RMW units
// (whole 25.6MB output is resident in the 192MB L2). The 320MB message
// stream uses the NT temporal hint so it does not evict the atomic-hot
// output lines.
// ---------------------------------------------------------------------------
__global__ __launch_bounds__(256) void SAGEStage3_ReduceSum_51994874085795_kernel(
        const float* __restrict__ msg,
        const long long* __restrict__ tgt,   // edge_index row 1 (int64)
        float* __restrict__ out,
        int num_edges) {
    unsigned gid = blockIdx.x * 256u + threadIdx.x;
    unsigned e = gid >> 4;               // edge id   (< 1.25M, fits u32)
    unsigned q = (gid & 15u) << 2;       // feature offset: 0,4,...,60
    if (e >= (unsigned)num_edges) return;

    // 16 consecutive lanes cover one 256B edge row -> perfectly coalesced.
    const v4f* mp = (const v4f*)(msg + (unsigned long long)e * FEAT + q);
    v4f m = __builtin_nontemporal_load(mp);   // global_load_b128, TH=NT

    unsigned t = (unsigned)tgt[e];            // 16 lanes share one cacheline
    // t*64+q < 6.4M -> u32 offset; base+u32 offset = GVS addressing
    float* op = out + (t << 6) + q;

    (void)__hip_atomic_fetch_add(op + 0, m.x, __ATOMIC_RELAXED, __HIP_MEMORY_SCOPE_AGENT);
    (void)__hip_atomic_fetch_add(op + 1, m.y, __ATOMIC_RELAXED, __HIP_MEMORY_SCOPE_AGENT);
    (void)__hip_atomic_fetch_add(op + 2, m.z, __ATOMIC_RELAXED, __HIP_MEMORY_SCOPE_AGENT);
    (void)__hip_atomic_fetch_add(op + 3, m.w, __ATOMIC_RELAXED, __HIP_MEMORY_SCOPE_AGENT);
}

// ---------------------------------------------------------------------------
// Zero-init the output (harness poisons d_out; atomics need a zero base).
// Regular temporal policy: these lines are about to be hammered by atomics,
// we WANT them resident in L2.
// ---------------------------------------------------------------------------
__global__ __launch_bounds__(256) void SAGE_zero_out_kernel(
        float* __restrict__ out, int n4) {
    int i = blockIdx.x * 256 + threadIdx.x;
    if (i < n4) {
        v4f z = {0.0f, 0.0f, 0.0f, 0.0f};
        ((v4f*)out)[i] = z;
    }
}

extern "C" void kernel_launch(void* const* d_in, const int* in_sizes, int n_in,
                              void* d_out, int out_size, void* d_ws, size_t ws_size,
                              hipStream_t stream) {
    const float*     msg = (const float*)d_in[0];          // (E, 64) f32
    const long long* ei  = (const long long*)d_in[1];      // (2, E) int64
    float*           out = (float*)d_out;                  // (N, 64) f32

    const int num_edges = in_sizes[0] / FEAT;
    const long long* tgt = ei + num_edges;                 // row 1 = targets

    // 1) zero the output
    int n4 = out_size / 4;
    int zblocks = (n4 + 255) / 256;
    SAGE_zero_out_kernel<<<zblocks, 256, 0, stream>>>(out, n4);

    // 2) scatter-add all edges (same stream -> ordered after the zero-fill)
    long long total_threads = (long long)num_edges * THREADS_PER_EDGE;
    int sblocks = (int)((total_threads + 255) / 256);
    SAGEStage3_ReduceSum_51994874085795_kernel<<<sblocks, 256, 0, stream>>>(
        msg, tgt, out, num_edges);
}